// FlowInitModule_35648228557120
// MI455X (gfx1250) — compile-verified
//
#include <hip/hip_runtime.h>
#include <hip/hip_bf16.h>

typedef __attribute__((ext_vector_type(2))) float v2f;
typedef __attribute__((ext_vector_type(8))) float v8f;

#define BB   2
#define NN   8192
#define CC   64
#define DD   64
#define WW   8
#define KK   16
#define EPS  1e-5f
#define SLOPE 0.1f

__device__ __forceinline__ float lrelu(float x) { return x >= 0.f ? x : SLOPE * x; }

// ---------------------------------------------------------------------------
// Kernel 1: KNN (full batch xyz staged in LDS: 96KB, fits 320KB CDNA5 LDS)
// writes knn_idx [B,N,K] and offsets [B,3,N,K]
// ---------------------------------------------------------------------------
__global__ __launch_bounds__(256) void knn_kernel(const float* __restrict__ xyz,
                                                  int* __restrict__ knn_idx,
                                                  float* __restrict__ offsets) {
    extern __shared__ char smem[];
    float* sx = (float*)smem;
    float* sy = sx + NN;
    float* sz = sy + NN;
    const int b   = blockIdx.y;
    const int tid = threadIdx.x;
    for (int i = tid; i < NN; i += 256) {
        sx[i] = xyz[(b * 3 + 0) * NN + i];
        sy[i] = xyz[(b * 3 + 1) * NN + i];
        sz[i] = xyz[(b * 3 + 2) * NN + i];
    }
    __syncthreads();

    const int n0 = blockIdx.x * 256 + tid;
    const float qx = sx[n0], qy = sy[n0], qz = sz[n0];
    const float sqi = qx * qx + qy * qy + qz * qz;

    float bd[KK];
    int   bi[KK];
#pragma unroll
    for (int s = 0; s < KK; ++s) { bd[s] = 3.4e38f; bi[s] = 0; }

    for (int j = 0; j < NN; ++j) {
        const float x = sx[j], y = sy[j], z = sz[j];
        const float sqj = x * x + y * y + z * z;
        const float dot = qx * x + qy * y + qz * z;
        const float d = sqi + sqj - 2.0f * dot;
        if (d < bd[KK - 1]) {
            bd[KK - 1] = d; bi[KK - 1] = j;
#pragma unroll
            for (int s = KK - 1; s > 0; --s) {
                if (bd[s] < bd[s - 1]) {
                    float td = bd[s]; bd[s] = bd[s - 1]; bd[s - 1] = td;
                    int   ti = bi[s]; bi[s] = bi[s - 1]; bi[s - 1] = ti;
                }
            }
        }
    }
#pragma unroll
    for (int s = 0; s < KK; ++s) {
        const int j = bi[s];
        knn_idx[((long)b * NN + n0) * KK + s] = j;
        offsets[(((long)b * 3 + 0) * NN + n0) * KK + s] = sx[j] - qx;
        offsets[(((long)b * 3 + 1) * NN + n0) * KK + s] = sy[j] - qy;
        offsets[(((long)b * 3 + 2) * NN + n0) * KK + s] = sz[j] - qz;
    }
}

// ---------------------------------------------------------------------------
// Kernel 2: WeightNet 3->8 pre-BN:  wnout[b,o,n,k]
// ---------------------------------------------------------------------------
__global__ __launch_bounds__(256) void weightnet_kernel(const float* __restrict__ offs,
                                                        const float* __restrict__ wn_w,
                                                        const float* __restrict__ wn_b,
                                                        float* __restrict__ wnout) {
    const long e = (long)blockIdx.x * 256 + threadIdx.x;     // [0, B*N*K)
    if (e >= (long)BB * NN * KK) return;
    const long b  = e / ((long)NN * KK);
    const long nk = e - b * (long)NN * KK;
    const float ox = offs[(b * 3 + 0) * (long)NN * KK + nk];
    const float oy = offs[(b * 3 + 1) * (long)NN * KK + nk];
    const float oz = offs[(b * 3 + 2) * (long)NN * KK + nk];
#pragma unroll
    for (int o = 0; o < WW; ++o) {
        wnout[(b * WW + o) * (long)NN * KK + nk] =
            wn_w[o * 3 + 0] * ox + wn_w[o * 3 + 1] * oy + wn_w[o * 3 + 2] * oz + wn_b[o];
    }
}

// ---------------------------------------------------------------------------
// BN stats (training mode, exact): mean + rsqrt(var+eps) per channel
// layout: x[b*bstride + c*per_b + i]
// ---------------------------------------------------------------------------
__global__ __launch_bounds__(256) void bn_stats_kernel(const float* __restrict__ x,
                                                       float* __restrict__ stats,
                                                       int nch, long per_b, long bstride, int nb) {
    __shared__ float ssum[256], ssq[256];
    const int c = blockIdx.x;
    const int tid = threadIdx.x;
    float s = 0.f, s2 = 0.f;
    for (int b = 0; b < nb; ++b) {
        const float* base = x + (long)b * bstride + (long)c * per_b;
        for (long i = tid; i < per_b; i += 256) {
            const float v = base[i];
            s += v; s2 += v * v;
        }
    }
    ssum[tid] = s; ssq[tid] = s2;
    __syncthreads();
    for (int o = 128; o > 0; o >>= 1) {
        if (tid < o) { ssum[tid] += ssum[tid + o]; ssq[tid] += ssq[tid + o]; }
        __syncthreads();
    }
    if (tid == 0) {
        const float cnt = (float)((long)nb * per_b);
        const float m = ssum[0] / cnt;
        const float var = ssq[0] / cnt - m * m;
        stats[c]       = m;
        stats[nch + c] = rsqrtf(var + EPS);
    }
}

// ---------------------------------------------------------------------------
// Kernel 3: PointConv. Block = 16 points of one batch.
//  Phase A: gather g=diff[:,idx] (64KB LDS) + BN+lrelu weightnet (8KB LDS),
//           compute agg[512 x 16] into LDS (32KB).
//  Phase B: linear 512->64 with V_WMMA_F32_16X16X4_F32 (4 waves x 128 WMMA).
//  writes pc [B,64,N] (pre-BN).
// ---------------------------------------------------------------------------
__global__ __launch_bounds__(256) void pointconv_kernel(const float* __restrict__ cost,
                                                        const float* __restrict__ points,
                                                        const int* __restrict__ knn_idx,
                                                        const float* __restrict__ wnout,
                                                        const float* __restrict__ wn_stats,
                                                        const float* __restrict__ wn_gamma,
                                                        const float* __restrict__ wn_beta,
                                                        const float* __restrict__ lin_w,
                                                        const float* __restrict__ lin_b,
                                                        float* __restrict__ pc) {
    extern __shared__ char smem[];
    int*   sIdx = (int*)smem;                       //  256 ints  =  1KB
    float* sW   = (float*)(smem + 1024);            // [p][o][k] 2048 f = 8KB
    float* sG   = (float*)(smem + 1024 + 8192);     // [p][k][c] 16384 f = 64KB
    float* sAgg = (float*)(smem + 1024 + 8192 + 65536); // [f][p] 8192 f = 32KB

    const int b   = blockIdx.y;
    const int n0  = blockIdx.x * 16;
    const int tid = threadIdx.x;

    // neighbor indices for the 16 points
    sIdx[tid] = knn_idx[((long)b * NN + n0 + (tid >> 4)) * KK + (tid & 15)];
    __syncthreads();

    // stage BN+lrelu'd weightnet output: sW[p][o][k]
    for (int t = tid; t < 16 * WW * KK; t += 256) {
        const int p = t >> 7, o = (t >> 4) & 7, k = t & 15;
        float x = wnout[(((long)b * WW + o) * NN + n0 + p) * KK + k];
        x = (x - wn_stats[o]) * wn_stats[WW + o] * wn_gamma[o] + wn_beta[o];
        sW[(p * WW + o) * KK + k] = lrelu(x);
    }
    // stage gathered diff: sG[p][k][c]
    for (int t = tid; t < 16 * KK * CC; t += 256) {
        const int p = t >> 10, k = (t >> 6) & 15, c = t & 63;
        const int j = sIdx[p * KK + k];
        const long a = ((long)b * CC + c) * NN + j;
        sG[(p * KK + k) * CC + c] = cost[a] - points[a];
    }
    __syncthreads();

    // Phase A2: agg[c*8+o][p] = sum_k g[c,k] * w[o,k]
    {
        const int p  = tid & 15;
        const int cb = (tid >> 4) * 4;
#pragma unroll
        for (int ci = 0; ci < 4; ++ci) {
            const int c = cb + ci;
            float acc[WW];
#pragma unroll
            for (int o = 0; o < WW; ++o) acc[o] = 0.f;
            for (int k = 0; k < KK; ++k) {
                const float gv = sG[(p * KK + k) * CC + c];
#pragma unroll
                for (int o = 0; o < WW; ++o) acc[o] += gv * sW[(p * WW + o) * KK + k];
            }
#pragma unroll
            for (int o = 0; o < WW; ++o) sAgg[(c * WW + o) * 16 + p] = acc[o];
        }
    }
    __syncthreads();

    // Phase B: pc[:, n0..n0+15] = lin_w[64x512] @ sAgg[512x16] via f32 WMMA
    const int wave = tid >> 5;
    if (wave < 4) {
        const int lane = tid & 31;
        const int half = lane >> 4;
        const int lid  = lane & 15;
        const int m0   = wave * 16;
        v8f acc = {};
        for (int kk = 0; kk < 512; kk += 4) {
            const int ka = kk + 2 * half;
            v2f a, bfr;
            a.x   = lin_w[(m0 + lid) * 512 + ka];
            a.y   = lin_w[(m0 + lid) * 512 + ka + 1];
            bfr.x = sAgg[ka * 16 + lid];
            bfr.y = sAgg[(ka + 1) * 16 + lid];
            acc = __builtin_amdgcn_wmma_f32_16x16x4_f32(false, a, false, bfr,
                                                        (short)0, acc, false, false);
        }
#pragma unroll
        for (int j = 0; j < 8; ++j) {
            const int m = m0 + j + 8 * half;
            pc[((long)b * CC + m) * NN + n0 + lid] = acc[j] + lin_b[m];
        }
    }
}

// ---------------------------------------------------------------------------
// Kernel 4: cost_mod = cost + lrelu(BN(pc))
// ---------------------------------------------------------------------------
__global__ __launch_bounds__(256) void costmod_kernel(const float* __restrict__ cost,
                                                      const float* __restrict__ pc,
                                                      const float* __restrict__ pc_stats,
                                                      const float* __restrict__ gamma,
                                                      const float* __restrict__ beta,
                                                      float* __restrict__ cost_mod) {
    const long e = (long)blockIdx.x * 256 + threadIdx.x;   // [0, B*C*N)
    if (e >= (long)BB * CC * NN) return;
    const int c = (int)((e / NN) % CC);
    float y = (pc[e] - pc_stats[c]) * pc_stats[CC + c] * gamma[c] + beta[c];
    cost_mod[e] = cost[e] + lrelu(y);
}

// ---------------------------------------------------------------------------
// Kernel 5: dense projection GEMM via V_WMMA_F32_16X16X4_F32.
//  Y[b, m, n] = sum_k Wm[m,k] * X[b, k, n] + bias[m]
//  grid = (L/16, M/16, B), block = 32 (one wave per 16x16 tile)
// ---------------------------------------------------------------------------
__global__ __launch_bounds__(32) void proj_gemm_wmma(const float* __restrict__ Wm,
                                                     const float* __restrict__ bias,
                                                     const float* __restrict__ X,
                                                     float* __restrict__ Y,
                                                     int Kd, int L, int M) {
    const int lane = threadIdx.x & 31;
    const int half = lane >> 4;
    const int lid  = lane & 15;
    const int n0   = blockIdx.x * 16;
    const int m0   = blockIdx.y * 16;
    const int b    = blockIdx.z;
    const float* Xb = X + (long)b * Kd * L;
    float*       Yb = Y + (long)b * M * L;

    v8f acc = {};
    for (int kk = 0; kk < Kd; kk += 4) {
        const int ka = kk + 2 * half;
        v2f a, bfr;
        a.x   = Wm[(m0 + lid) * Kd + ka];
        a.y   = Wm[(m0 + lid) * Kd + ka + 1];
        bfr.x = Xb[(long)ka * L + n0 + lid];
        bfr.y = Xb[(long)(ka + 1) * L + n0 + lid];
        acc = __builtin_amdgcn_wmma_f32_16x16x4_f32(false, a, false, bfr,
                                                    (short)0, acc, false, false);
    }
#pragma unroll
    for (int j = 0; j < 8; ++j) {
        const int m = m0 + j + 8 * half;
        Yb[(long)m * L + n0 + lid] = acc[j] + bias[m];
    }
}

// ---------------------------------------------------------------------------
// Kernel 6: fused attention. One wave32 per point; 2 channels per lane.
//  k_/v of gathered columns == gather of projected columns (1x1 conv).
// ---------------------------------------------------------------------------
__global__ __launch_bounds__(256) void attn_kernel(const float* __restrict__ qb,
                                                   const float* __restrict__ kb,
                                                   const float* __restrict__ vb,
                                                   const int* __restrict__ knn_idx,
                                                   const float* __restrict__ offs,
                                                   const float* __restrict__ wpos,
                                                   const float* __restrict__ bpos,
                                                   float* __restrict__ out) {
    __shared__ float sV[8][DD][KK];          // 32KB: (v + pos) per wave
    const int wave = threadIdx.x >> 5;
    const int lane = threadIdx.x & 31;
    const long pidx = (long)blockIdx.x * 8 + wave;   // [0, B*N)
    const int b = (int)(pidx / NN);
    const int n = (int)(pidx % NN);
    const int d0 = lane * 2, d1 = lane * 2 + 1;

    const float q0 = qb[((long)b * DD + d0) * NN + n];
    const float q1 = qb[((long)b * DD + d1) * NN + n];
    const float w0x = wpos[d0 * 3], w0y = wpos[d0 * 3 + 1], w0z = wpos[d0 * 3 + 2];
    const float w1x = wpos[d1 * 3], w1y = wpos[d1 * 3 + 1], w1z = wpos[d1 * 3 + 2];
    const float bp0 = bpos[d0], bp1 = bpos[d1];

    float sc[KK];
#pragma unroll
    for (int k = 0; k < KK; ++k) {
        const int j = knn_idx[((long)b * NN + n) * KK + k];
        const float ox = offs[(((long)b * 3 + 0) * NN + n) * KK + k];
        const float oy = offs[(((long)b * 3 + 1) * NN + n) * KK + k];
        const float oz = offs[(((long)b * 3 + 2) * NN + n) * KK + k];
        const float p0 = w0x * ox + w0y * oy + w0z * oz + bp0;
        const float p1 = w1x * ox + w1y * oy + w1z * oz + bp1;
        const float k0 = kb[((long)b * DD + d0) * NN + j];
        const float k1 = kb[((long)b * DD + d1) * NN + j];
        const float v0 = vb[((long)b * DD + d0) * NN + j];
        const float v1 = vb[((long)b * DD + d1) * NN + j];
        sV[wave][d0][k] = v0 + p0;
        sV[wave][d1][k] = v1 + p1;
        float s = q0 * (k0 + p0) + q1 * (k1 + p1);
#pragma unroll
        for (int o = 16; o > 0; o >>= 1) s += __shfl_xor(s, o, 32);
        sc[k] = s * 0.125f;                          // 1/sqrt(64)
    }
    float m = sc[0];
#pragma unroll
    for (int k = 1; k < KK; ++k) m = fmaxf(m, sc[k]);
    float tot = 0.f, ek[KK];
#pragma unroll
    for (int k = 0; k < KK; ++k) { ek[k] = expf(sc[k] - m); tot += ek[k]; }
    const float inv = 1.0f / tot;
    float o0 = 0.f, o1 = 0.f;
#pragma unroll
    for (int k = 0; k < KK; ++k) {
        const float a = ek[k] * inv;
        o0 += a * sV[wave][d0][k];
        o1 += a * sV[wave][d1][k];
    }
    out[((long)b * DD + d0) * NN + n] = o0;
    out[((long)b * DD + d1) * NN + n] = o1;
}

// ---------------------------------------------------------------------------
extern "C" void kernel_launch(void* const* d_in, const int* in_sizes, int n_in,
                              void* d_out, int out_size, void* d_ws, size_t ws_size,
                              hipStream_t stream) {
    const float* xyz      = (const float*)d_in[0];
    const float* points   = (const float*)d_in[1];
    const float* cost     = (const float*)d_in[2];
    const float* wn_w     = (const float*)d_in[3];
    const float* wn_b     = (const float*)d_in[4];
    const float* wn_gamma = (const float*)d_in[5];
    const float* wn_beta  = (const float*)d_in[6];
    const float* lin_w    = (const float*)d_in[7];
    const float* lin_b    = (const float*)d_in[8];
    const float* bn_gamma = (const float*)d_in[9];
    const float* bn_beta  = (const float*)d_in[10];
    const float* wq       = (const float*)d_in[11];
    const float* bq       = (const float*)d_in[12];
    const float* wk       = (const float*)d_in[13];
    const float* bk       = (const float*)d_in[14];
    const float* wv       = (const float*)d_in[15];
    const float* bv       = (const float*)d_in[16];
    const float* wpos     = (const float*)d_in[17];
    const float* bpos     = (const float*)d_in[18];
    float* out = (float*)d_out;

    char* wp = (char*)d_ws;
    auto alloc = [&](size_t bytes) -> void* {
        void* p = (void*)wp;
        wp += (bytes + 255) & ~(size_t)255;
        return p;
    };
    int*   knn      = (int*)  alloc(sizeof(int)   * BB * NN * KK);      //  1MB
    float* offs     = (float*)alloc(sizeof(float) * BB * 3 * NN * KK);  //  3MB
    float* wnout    = (float*)alloc(sizeof(float) * BB * WW * NN * KK); //  8MB
    float* wn_stats = (float*)alloc(sizeof(float) * 2 * WW);
    float* pc       = (float*)alloc(sizeof(float) * BB * CC * NN);      //  4MB
    float* pc_stats = (float*)alloc(sizeof(float) * 2 * CC);
    float* cost_mod = (float*)alloc(sizeof(float) * BB * CC * NN);      //  4MB
    float* qbuf     = (float*)alloc(sizeof(float) * BB * DD * NN);      //  4MB
    float* kbuf     = (float*)alloc(sizeof(float) * BB * DD * NN);      //  4MB
    float* vbuf     = (float*)alloc(sizeof(float) * BB * DD * NN);      //  4MB

    // 1) KNN + offsets (96KB LDS)
    knn_kernel<<<dim3(NN / 256, BB), 256, 3 * NN * sizeof(float), stream>>>(xyz, knn, offs);

    // 2) WeightNet pre-BN
    weightnet_kernel<<<(BB * NN * KK) / 256, 256, 0, stream>>>(offs, wn_w, wn_b, wnout);

    // 3) BN stats for weightnet (8 channels over B*N*K)
    bn_stats_kernel<<<WW, 256, 0, stream>>>(wnout, wn_stats, WW,
                                            (long)NN * KK, (long)WW * NN * KK, BB);

    // 4) PointConv (gather + agg + WMMA linear); 105KB dynamic LDS
    pointconv_kernel<<<dim3(NN / 16, BB), 256, 1024 + 8192 + 65536 + 32768, stream>>>(
        cost, points, knn, wnout, wn_stats, wn_gamma, wn_beta, lin_w, lin_b, pc);

    // 5) BN stats for pc (64 channels over B*N)
    bn_stats_kernel<<<CC, 256, 0, stream>>>(pc, pc_stats, CC, (long)NN, (long)CC * NN, BB);

    // 6) cost_mod = cost + lrelu(BN(pc))
    costmod_kernel<<<(BB * CC * NN) / 256, 256, 0, stream>>>(cost, pc, pc_stats,
                                                             bn_gamma, bn_beta, cost_mod);

    // 7) q/k/v projections (WMMA f32 GEMMs, per batch)
    proj_gemm_wmma<<<dim3(NN / 16, DD / 16, BB), 32, 0, stream>>>(wq, bq, cost_mod, qbuf, CC, NN, DD);
    proj_gemm_wmma<<<dim3(NN / 16, DD / 16, BB), 32, 0, stream>>>(wk, bk, cost_mod, kbuf, CC, NN, DD);
    proj_gemm_wmma<<<dim3(NN / 16, DD / 16, BB), 32, 0, stream>>>(wv, bv, cost_mod, vbuf, CC, NN, DD);

    // 8) fused attention (one wave32 per point)
    attn_kernel<<<(BB * NN) / 8, 256, 0, stream>>>(qbuf, kbuf, vbuf, knn, offs,
                                                   wpos, bpos, out);
}